// MultiHeadSelfAttention_8770323219237
// MI455X (gfx1250) — compile-verified
//
#include <hip/hip_runtime.h>

// MI455X / gfx1250, wave32. fp32 WMMA (V_WMMA_F32_16X16X4_F32) everywhere to
// match the fp32 reference numerics; problem is HBM-bound on the 403MB attn
// matrix, so fp32 matrix math is the right precision/speed point.

#define B_   2
#define T_   2048
#define D_   768
#define H_   12
#define DH_  64
#define NROW (B_ * T_)            // 4096
#define BHTD (B_ * H_ * T_ * DH_) // 3145728

typedef __attribute__((ext_vector_type(2))) float v2f;
typedef __attribute__((ext_vector_type(4))) float v4f;
typedef __attribute__((ext_vector_type(8))) float v8f;

static __device__ __forceinline__ v8f wmma_f32(v2f a, v2f b, v8f c) {
    // 8 args: (neg_a, A, neg_b, B, c_mod, C, reuse_a, reuse_b)
    return __builtin_amdgcn_wmma_f32_16x16x4_f32(false, a, false, b, (short)0, c, false, false);
}

// ---------------------------------------------------------------------------
// Kernel 1: fused QKV projection.  y = x @ W^T + b for W in {Wq,Wk,Wv}.
// Block tile: 128(M) x 64(N), K staged via LDS in chunks of 32.
// Output written head-major: ws[mat][b][h][t][e].
// ---------------------------------------------------------------------------
__global__ __launch_bounds__(256) void qkv_kernel(
    const float* __restrict__ x,
    const float* __restrict__ Wq, const float* __restrict__ bq,
    const float* __restrict__ Wk, const float* __restrict__ bk,
    const float* __restrict__ Wv, const float* __restrict__ bv,
    float* __restrict__ qkv_ws)
{
    __shared__ float xs[128 * 33];   // x tile, padded stride (64-bank friendly)
    __shared__ float wsd[64 * 33];   // W tile

    const int bid    = blockIdx.x;
    const int nchunk = bid % 12;          // 12 x 64 over D
    const int mblk   = (bid / 12) % 32;   // 32 x 128 over 4096 rows
    const int mat    = bid / (12 * 32);   // 0=Q 1=K 2=V

    const float* W    = (mat == 0) ? Wq : (mat == 1) ? Wk : Wv;
    const float* bias = (mat == 0) ? bq : (mat == 1) ? bk : bv;
    float* dst = qkv_ws + (size_t)mat * BHTD;

    const int m0 = mblk * 128, n0 = nchunk * 64;
    const int t    = threadIdx.x;
    const int wave = t >> 5, lane = t & 31;
    const int lm = lane & 15, kh = lane >> 4;   // M/N index, K-pair select
    const int mr = wave * 16;                   // wave's M sub-tile

    v8f acc[4] = {};

    const int rq = t >> 3;        // 0..31 staging row
    const int kq = (t & 7) * 4;   // 0..28 staging col (float4)

    for (int k0 = 0; k0 < D_; k0 += 32) {
        #pragma unroll
        for (int rr = 0; rr < 4; ++rr) {           // x: 128 x 32
            int row = rq + rr * 32;
            v4f vx = *(const v4f*)(x + (size_t)(m0 + row) * D_ + k0 + kq);
            xs[row * 33 + kq + 0] = vx.x;
            xs[row * 33 + kq + 1] = vx.y;
            xs[row * 33 + kq + 2] = vx.z;
            xs[row * 33 + kq + 3] = vx.w;
        }
        #pragma unroll
        for (int rr = 0; rr < 2; ++rr) {           // W: 64 x 32
            int row = rq + rr * 32;
            v4f vw = *(const v4f*)(W + (size_t)(n0 + row) * D_ + k0 + kq);
            wsd[row * 33 + kq + 0] = vw.x;
            wsd[row * 33 + kq + 1] = vw.y;
            wsd[row * 33 + kq + 2] = vw.z;
            wsd[row * 33 + kq + 3] = vw.w;
        }
        __syncthreads();
        #pragma unroll
        for (int kk = 0; kk < 32; kk += 4) {
            v2f a;
            a.x = xs[(mr + lm) * 33 + kk + 2 * kh + 0];
            a.y = xs[(mr + lm) * 33 + kk + 2 * kh + 1];
            #pragma unroll
            for (int j = 0; j < 4; ++j) {
                v2f b;
                b.x = wsd[(j * 16 + lm) * 33 + kk + 2 * kh + 0];
                b.y = wsd[(j * 16 + lm) * 33 + kk + 2 * kh + 1];
                acc[j] = wmma_f32(a, b, acc[j]);
            }
        }
        __syncthreads();
    }

    #pragma unroll
    for (int j = 0; j < 4; ++j) {
        int ng = n0 + j * 16 + lm;           // output feature
        int h = ng >> 6, e = ng & 63;
        float bb = bias[ng];
        #pragma unroll
        for (int r = 0; r < 8; ++r) {
            int mg = m0 + mr + r + 8 * kh;   // global row (b*T + t)
            int b = mg >> 11, tt = mg & 2047;
            dst[((size_t)(b * H_ + h) * T_ + tt) * DH_ + e] = acc[j][r] + bb;
        }
    }
}

// ---------------------------------------------------------------------------
// Kernel 2a: S = Q K^T / sqrt(Dh) + causal mask (pre-softmax), stored into the
// attn output region.  Wave tile 16x64; fully masked tiles skipped (uniform).
// ---------------------------------------------------------------------------
__global__ __launch_bounds__(256) void scores_kernel(
    const float* __restrict__ qkv_ws, float* __restrict__ attn)
{
    const int wid  = blockIdx.x * 8 + (threadIdx.x >> 5);
    const int lane = threadIdx.x & 31;
    const int ct = wid & 31;            // 32 col tiles of 64
    const int mt = (wid >> 5) & 127;    // 128 row tiles of 16
    const int bh = wid >> 12;           // 0..23
    const int r0 = mt * 16, c0 = ct * 64;
    if (c0 > r0 + 15) return;           // wave-uniform: entire tile masked

    const float* Q = qkv_ws + (size_t)bh * T_ * DH_;
    const float* K = qkv_ws + (size_t)BHTD + (size_t)bh * T_ * DH_;
    const int lm = lane & 15, kh = lane >> 4;

    v8f acc[4] = {};
    #pragma unroll
    for (int d = 0; d < DH_; d += 4) {
        v2f a = *(const v2f*)(Q + (size_t)(r0 + lm) * DH_ + d + 2 * kh);
        #pragma unroll
        for (int j = 0; j < 4; ++j) {
            v2f b = *(const v2f*)(K + (size_t)(c0 + j * 16 + lm) * DH_ + d + 2 * kh);
            acc[j] = wmma_f32(a, b, acc[j]);
        }
    }

    float* Sp = attn + (size_t)bh * T_ * T_;
    const float scale = 0.125f;   // 1/sqrt(64)
    #pragma unroll
    for (int j = 0; j < 4; ++j) {
        int col = c0 + j * 16 + lm;
        #pragma unroll
        for (int r = 0; r < 8; ++r) {
            int row = r0 + r + 8 * kh;
            float v = acc[j][r] * scale + (col > row ? -1e9f : 0.0f);
            Sp[(size_t)row * T_ + col] = v;
        }
    }
}

// ---------------------------------------------------------------------------
// Kernel 2b: row softmax in place.  One wave per row; online max/sum over the
// i+1 causal-valid columns, shuffle reduction, zeros for the masked tail.
// ---------------------------------------------------------------------------
__global__ __launch_bounds__(256) void softmax_kernel(float* __restrict__ attn)
{
    const int row  = blockIdx.x * 8 + (threadIdx.x >> 5);
    const int lane = threadIdx.x & 31;
    const int bh = row >> 11;
    const int i  = row & 2047;
    float* p = attn + (size_t)bh * T_ * T_ + (size_t)i * T_;
    const int L = i + 1;

    float m = -1e30f, s = 0.0f;
    for (int j = lane; j < L; j += 32) {
        float xv = p[j];
        float mn = fmaxf(m, xv);
        s = s * __expf(m - mn) + __expf(xv - mn);
        m = mn;
    }
    #pragma unroll
    for (int off = 16; off >= 1; off >>= 1) {
        float m2 = __shfl_xor(m, off, 32);
        float s2 = __shfl_xor(s, off, 32);
        float mn = fmaxf(m, m2);
        s = s * __expf(m - mn) + s2 * __expf(m2 - mn);
        m = mn;
    }
    float inv = 1.0f / s;
    for (int j = lane; j < L; j += 32) p[j] = __expf(p[j] - m) * inv;
    for (int j = L + lane; j < T_; j += 32) p[j] = 0.0f;   // exact zeros (masked)
}

// ---------------------------------------------------------------------------
// Kernel 2c: O = P V.  Wave tile 16 x 64(=Dh); K-loop truncated at causal
// diagonal (P is zero beyond it).  Output [B,T,H,Dh].
// ---------------------------------------------------------------------------
__global__ __launch_bounds__(256) void av_kernel(
    const float* __restrict__ qkv_ws, const float* __restrict__ attn,
    float* __restrict__ heads)
{
    const int wid  = blockIdx.x * 8 + (threadIdx.x >> 5);
    const int lane = threadIdx.x & 31;
    const int mt = wid & 127;
    const int bh = wid >> 7;    // 0..23
    const int r0 = mt * 16;

    const float* P = attn + (size_t)bh * T_ * T_;
    const float* V = qkv_ws + 2 * (size_t)BHTD + (size_t)bh * T_ * DH_;
    const int lm = lane & 15, kh = lane >> 4;

    v8f acc[4] = {};
    const int kmax = r0 + 16;   // causal: P[row][k]==0 for k>row
    for (int k = 0; k < kmax; k += 4) {
        v2f a = *(const v2f*)(P + (size_t)(r0 + lm) * T_ + k + 2 * kh);
        #pragma unroll
        for (int j = 0; j < 4; ++j) {
            int n = j * 16 + lm;
            v2f b;
            b.x = V[(size_t)(k + 2 * kh + 0) * DH_ + n];
            b.y = V[(size_t)(k + 2 * kh + 1) * DH_ + n];
            acc[j] = wmma_f32(a, b, acc[j]);
        }
    }

    const int b = bh / H_, h = bh % H_;
    #pragma unroll
    for (int j = 0; j < 4; ++j) {
        int e = j * 16 + lm;
        #pragma unroll
        for (int r = 0; r < 8; ++r) {
            int tt = r0 + r + 8 * kh;
            heads[(((size_t)b * T_ + tt) * H_ + h) * DH_ + e] = acc[j][r];
        }
    }
}

extern "C" void kernel_launch(void* const* d_in, const int* in_sizes, int n_in,
                              void* d_out, int out_size, void* d_ws, size_t ws_size,
                              hipStream_t stream)
{
    (void)in_sizes; (void)n_in; (void)out_size; (void)ws_size;
    const float* x  = (const float*)d_in[0];
    const float* Wq = (const float*)d_in[1];
    const float* bq = (const float*)d_in[2];
    const float* Wk = (const float*)d_in[3];
    const float* bk = (const float*)d_in[4];
    const float* Wv = (const float*)d_in[5];
    const float* bv = (const float*)d_in[6];
    // d_in[7] = attn_mask: causal structure handled analytically.

    float* heads = (float*)d_out;                  // [B,T,H,Dh]
    float* attn  = (float*)d_out + (size_t)BHTD;   // [B,H,T,T]
    float* qkv   = (float*)d_ws;                   // 3 * BHTD floats (~37.7 MB)

    qkv_kernel    <<<1152, 256, 0, stream>>>(x, Wq, bq, Wk, bk, Wv, bv, qkv);
    scores_kernel <<<12288, 256, 0, stream>>>(qkv, attn);
    softmax_kernel<<<6144, 256, 0, stream>>>(attn);
    av_kernel     <<<384,  256, 0, stream>>>(qkv, attn, heads);
}